// AutomatonNetwork_75127567941932
// MI455X (gfx1250) — compile-verified
//
#include <hip/hip_runtime.h>

// AutomatonNetwork walk on gfx1250 (MI455X).
// Sequential chain of 4096 matvecs (512-vec @ 512x512 f32), done by ONE
// workgroup of 32 wave32 waves. Each wave computes a 16-column tile of the
// new state vector with V_WMMA_F32_16X16X4_F32 (vector in row 0 of A).
// 4 independent accumulator chains break the WMMA RAW dependency. A-row-0
// masking is done by ADDRESS selection (non-head lanes read a zero LDS pad),
// so the k-loop is pure ds_load_2addr_b64 + global_load_b32 + v_wmma.
// transfer[] (128 MB) is L2-resident (192 MB L2); next step's matrix is
// prefetched with global_prefetch_b8 while the current step computes.

typedef __attribute__((ext_vector_type(2))) float v2f;
typedef __attribute__((ext_vector_type(8))) float v8f;

#define NSTATE 512
#define LTOK   4096
#define BLOCK  1024   // 32 waves of 32

__device__ __forceinline__ float wave_sum(float v) {
  v += __shfl_down(v, 16, 32);
  v += __shfl_down(v, 8, 32);
  v += __shfl_down(v, 4, 32);
  v += __shfl_down(v, 2, 32);
  v += __shfl_down(v, 1, 32);
  return v;
}

__global__ __launch_bounds__(BLOCK)
void automaton_walk(const int* __restrict__ tokens,
                    const float* __restrict__ start,
                    const float* __restrict__ transfer,
                    const float* __restrict__ probs,
                    const float* __restrict__ finals,
                    float* __restrict__ out) {
  __shared__ float vbuf[2][NSTATE];   // double-buffered state vector
  __shared__ float zpad[NSTATE];      // permanent zeros: A rows 1..15 source
  __shared__ float part[BLOCK / 32];  // per-wave reduction partials
  __shared__ float probAcc;

  const int tid  = threadIdx.x;
  const int lane = tid & 31;
  const int wave = tid >> 5;

  if (tid < NSTATE) {
    vbuf[0][tid] = start[tid];
    zpad[tid] = 0.0f;
  }
  if (tid == 0) probAcc = 1.0f;
  __syncthreads();

  // WMMA tile assignment: wave w owns output columns [16w, 16w+16).
  const int  j0      = wave * 16;
  const int  colBase = j0 + (lane & 15);
  const int  krow    = (lane >> 4) * 2;     // lanes 0-15: K rows 0/1, lanes 16-31: K rows 2/3
  const bool isHead  = ((lane & 15) == 0);  // lanes 0 and 16 carry A's row 0

  for (int t = 0; t < LTOK; ++t) {
    const float* cur = vbuf[t & 1];
    float*       nxt = vbuf[(t & 1) ^ 1];
    const int    tok = tokens[t];
    const float* Tm  = transfer + (size_t)tok * (NSTATE * NSTATE);

    // --- emission: probAcc *= dot(cur, probs[tok]) (uses PRE-update vector) ---
    float ev = (tid < NSTATE) ? cur[tid] * probs[(size_t)tok * NSTATE + tid] : 0.0f;
    float ws = wave_sum(ev);
    if (lane == 0) part[wave] = ws;
    __syncthreads();
    if (tid < 32) {
      float s = wave_sum(part[tid]);
      if (tid == 0) probAcc *= s;
    }

    // A source: head lanes walk the live vector, all others walk zeros.
    const float* asrc = isHead ? cur : zpad;

    // --- matvec: nxt = cur @ Tm via f32 WMMA over K in chunks of 4,
    //     4 interleaved accumulator chains for XDL-pipe ILP ---
    v8f acc0 = {}, acc1 = {}, acc2 = {}, acc3 = {};
    for (int k = 0; k < NSTATE; k += 16) {
      v2f a[4], b[4];
#pragma unroll
      for (int u = 0; u < 4; ++u) {
        const int kb = k + 4 * u;
        a[u] = *(const v2f*)(asrc + kb + krow);  // broadcast LDS read
        // B(4x16): rows kb+krow, kb+krow+1 at this lane's column.
        b[u].x = Tm[(size_t)(kb + krow) * NSTATE + colBase];
        b[u].y = Tm[(size_t)(kb + krow + 1) * NSTATE + colBase];
      }
      acc0 = __builtin_amdgcn_wmma_f32_16x16x4_f32(false, a[0], false, b[0],
                                                   (short)0, acc0, false, false);
      acc1 = __builtin_amdgcn_wmma_f32_16x16x4_f32(false, a[1], false, b[1],
                                                   (short)0, acc1, false, false);
      acc2 = __builtin_amdgcn_wmma_f32_16x16x4_f32(false, a[2], false, b[2],
                                                   (short)0, acc2, false, false);
      acc3 = __builtin_amdgcn_wmma_f32_16x16x4_f32(false, a[3], false, b[3],
                                                   (short)0, acc3, false, false);
    }
    v8f accA = acc0 + acc1;
    v8f accB = acc2 + acc3;
    v8f acc  = accA + accB;
    if (lane < 16) nxt[j0 + lane] = acc[0];  // D row 0 = new vector slice

    // --- prefetch next symbol's 1MB matrix into cache (overlaps this step) ---
    if (t + 1 < LTOK) {
      const float* Tn = transfer + (size_t)tokens[t + 1] * (NSTATE * NSTATE);
      for (int q = tid; q < (NSTATE * NSTATE) / 16; q += BLOCK)
        __builtin_prefetch(Tn + (size_t)q * 16, 0, 1);
    }
    __syncthreads();  // commit nxt; also fences reuse of `part` and old `cur`
  }

  // --- finalize: probAcc *= dot(internal, finals) ---
  const float* fin = vbuf[LTOK & 1];
  float fv = (tid < NSTATE) ? fin[tid] * finals[tid] : 0.0f;
  float ws = wave_sum(fv);
  if (lane == 0) part[wave] = ws;
  __syncthreads();
  if (tid < 32) {
    float s = wave_sum(part[tid]);
    if (tid == 0) probAcc *= s;
  }
  __syncthreads();

  if (tid < NSTATE) out[tid] = fin[tid];
  if (tid == 0) out[NSTATE] = probAcc;
}

extern "C" void kernel_launch(void* const* d_in, const int* in_sizes, int n_in,
                              void* d_out, int out_size, void* d_ws, size_t ws_size,
                              hipStream_t stream) {
  (void)in_sizes; (void)n_in; (void)out_size; (void)d_ws; (void)ws_size;
  const int*   tokens   = (const int*)d_in[0];
  const float* start    = (const float*)d_in[1];
  const float* transfer = (const float*)d_in[2];
  const float* probs    = (const float*)d_in[3];
  const float* finals   = (const float*)d_in[4];
  float*       out      = (float*)d_out;

  automaton_walk<<<1, BLOCK, 0, stream>>>(tokens, start, transfer, probs, finals, out);
}